// MySGConv_11622181503641
// MI455X (gfx1250) — compile-verified
//
#include <hip/hip_runtime.h>

typedef __attribute__((ext_vector_type(2))) float v2f;
typedef __attribute__((ext_vector_type(8))) float v8f;

#define DFEAT 128
#define LDSW  132   // padded LDS row stride (floats) to avoid bank conflicts

// ---------------- zero workspace ----------------
__global__ void zero_f32(float* __restrict__ p, int n) {
    int i = blockIdx.x * blockDim.x + threadIdx.x;
    int stride = gridDim.x * blockDim.x;
    for (; i < n; i += stride) p[i] = 0.0f;
}

// ---------------- degree count ----------------
__global__ void deg_count(const int* __restrict__ dst, float* __restrict__ deg, int ne) {
    int e = blockIdx.x * blockDim.x + threadIdx.x;
    if (e < ne) atomicAdd(&deg[dst[e]], 1.0f);
}

// ---------------- deg -> rsqrt(max(deg,1)) in place ----------------
__global__ void deg_to_norm(float* __restrict__ deg, int n) {
    int i = blockIdx.x * blockDim.x + threadIdx.x;
    if (i < n) {
        float d = deg[i];
        d = d < 1.0f ? 1.0f : d;
        deg[i] = rsqrtf(d);
    }
}

// ---------------- SpMM scatter: agg[dst] += x[src] * norm[src] ----------------
// one wave32 per edge; lane l moves x[src][4l..4l+3] (float4 -> 512B/wave, coalesced)
__global__ __launch_bounds__(256) void spmm_scatter(
    const float* __restrict__ x, const int* __restrict__ src,
    const int* __restrict__ dst, const float* __restrict__ norm,
    float* __restrict__ agg, int ne)
{
    int lane = threadIdx.x & 31;
    int e = blockIdx.x * (blockDim.x >> 5) + (threadIdx.x >> 5);
    if (e >= ne) return;
    int s = src[e];
    int d = dst[e];
    float ns = norm[s];
    float4 v = ((const float4*)(x + (size_t)s * DFEAT))[lane];
    float* o = agg + (size_t)d * DFEAT + lane * 4;
    atomicAdd(o + 0, v.x * ns);
    atomicAdd(o + 1, v.y * ns);
    atomicAdd(o + 2, v.z * ns);
    atomicAdd(o + 3, v.w * ns);
}

// ---------------- fused: relu((agg*norm) @ Wc^T + bc) @ Wl^T + bl, relu ----------------
// block = 256 threads = 8 waves; block handles 16 rows; wave w owns cols [16w,16w+16)
// fp32 WMMA 16x16x4, K=128 -> 32 wmma per tile per GEMM
__global__ __launch_bounds__(256) void sgconv_fused_gemm(
    const float* __restrict__ agg, const float* __restrict__ norm,
    const float* __restrict__ wc, const float* __restrict__ bc,
    const float* __restrict__ wl, const float* __restrict__ bl,
    float* __restrict__ out)
{
    __shared__ float shA[16 * LDSW];  // staged input tile (agg * norm[dst-row])
    __shared__ float shB[16 * LDSW];  // intermediate after GEMM1+bias+relu

    const int t  = threadIdx.x;
    const int r0 = blockIdx.x * 16;   // 50000 % 16 == 0 -> no tail

    // stage 16x128 tile, scaled by norm of each row (the D^{-1/2} on the left)
    for (int i = t; i < 16 * DFEAT; i += 256) {
        int row = i >> 7;
        int col = i & (DFEAT - 1);
        shA[row * LDSW + col] = agg[(size_t)(r0 + row) * DFEAT + col] * norm[r0 + row];
    }
    __syncthreads();

    const int wave = t >> 5;
    const int lane = t & 31;
    const int kh   = lane >> 4;    // K half-select (ISA A/B fragment layout)
    const int nn   = lane & 15;    // A row m / B col n
    const int n0   = wave * 16;

    // ---- GEMM1: shA(16x128) @ wc^T -> 16x16 tile per wave ----
    v8f acc = {};
#pragma unroll
    for (int kk = 0; kk < 32; ++kk) {
        int ko = kk * 4 + 2 * kh;
        v2f a = *(const v2f*)(shA + nn * LDSW + ko);                 // A[m][k,k+1]
        v2f b = *(const v2f*)(wc + (size_t)(n0 + nn) * DFEAT + ko);  // B[k][n] = wc[n][k]
        acc = __builtin_amdgcn_wmma_f32_16x16x4_f32(
            false, a, false, b, (short)0, acc, false, false);
    }
    // bias + relu -> shB (C/D layout: vgpr r -> row r + 8*kh, col = nn)
    {
        float bb = bc[n0 + nn];
#pragma unroll
        for (int r = 0; r < 8; ++r) {
            float v = acc[r] + bb;
            shB[(r + 8 * kh) * LDSW + n0 + nn] = v > 0.0f ? v : 0.0f;
        }
    }
    __syncthreads();

    // ---- GEMM2: shB(16x128) @ wl^T -> out, bias + relu ----
    v8f acc2 = {};
#pragma unroll
    for (int kk = 0; kk < 32; ++kk) {
        int ko = kk * 4 + 2 * kh;
        v2f a = *(const v2f*)(shB + nn * LDSW + ko);
        v2f b = *(const v2f*)(wl + (size_t)(n0 + nn) * DFEAT + ko);
        acc2 = __builtin_amdgcn_wmma_f32_16x16x4_f32(
            false, a, false, b, (short)0, acc2, false, false);
    }
    {
        float bb = bl[n0 + nn];
#pragma unroll
        for (int r = 0; r < 8; ++r) {
            float v = acc2[r] + bb;
            out[(size_t)(r0 + r + 8 * kh) * DFEAT + n0 + nn] = v > 0.0f ? v : 0.0f;
        }
    }
}

extern "C" void kernel_launch(void* const* d_in, const int* in_sizes, int n_in,
                              void* d_out, int out_size, void* d_ws, size_t ws_size,
                              hipStream_t stream) {
    const float* x   = (const float*)d_in[0];
    const int*   src = (const int*)d_in[1];
    const int*   dst = (const int*)d_in[2];
    const float* wc  = (const float*)d_in[3];
    const float* bc  = (const float*)d_in[4];
    const float* wl  = (const float*)d_in[5];
    const float* bl  = (const float*)d_in[6];
    float* out = (float*)d_out;

    const int n_nodes = in_sizes[0] / DFEAT;   // 50000
    const int n_edges = in_sizes[1];           // 800000

    // workspace: [norm (n_nodes, padded)] [agg (n_nodes*128)]
    size_t norm_cap = ((size_t)n_nodes + 255) & ~(size_t)255;
    float* norm = (float*)d_ws;
    float* agg  = norm + norm_cap;
    int n_zero = (int)(norm_cap + (size_t)n_nodes * DFEAT);

    // 1) zero workspace (norm/deg + agg) every call
    zero_f32<<<2048, 256, 0, stream>>>((float*)d_ws, n_zero);
    // 2) in-degree via float atomics
    deg_count<<<(n_edges + 255) / 256, 256, 0, stream>>>(dst, norm, n_edges);
    // 3) deg -> rsqrt(max(deg,1))
    deg_to_norm<<<(n_nodes + 255) / 256, 256, 0, stream>>>(norm, n_nodes);
    // 4) scatter-add SpMM: one wave per edge (8 edges / 256-thread block)
    spmm_scatter<<<(n_edges + 7) / 8, 256, 0, stream>>>(x, src, dst, norm, agg, n_edges);
    // 5) fused norm-scale + GEMM1(relu) + GEMM2(relu) with fp32 WMMA
    sgconv_fused_gemm<<<n_nodes / 16, 256, 0, stream>>>(agg, norm, wc, bc, wl, bl, out);
}